// HDCFactMemory_59863254172677
// MI455X (gfx1250) — compile-verified
//
#include <hip/hip_runtime.h>
#include <math.h>

// ---------------------------------------------------------------------------
// HDC fact memory, MI455X (gfx1250, wave32).
//   B=2, T=4096, D=4096, f32 in / f32 out.
// Memory-bound: 256 MiB min traffic => ~11us at 23.3 TB/s. Gate GEMVs done
// with V_WMMA_F32_16X16X4_F32 (exact f32 accumulation, 4 independent
// accumulator chains), cumsum via chunked 3-phase scan for parallelism.
// ---------------------------------------------------------------------------

namespace {
constexpr int kB = 2;
constexpr int kT = 4096;
constexpr int kD = 4096;
constexpr int kNChunk = 32;
constexpr int kChunk = kT / kNChunk;   // 128
constexpr int kRows = kB * kT;         // 8192 gate rows
}

typedef __attribute__((ext_vector_type(2))) float v2f;
typedef __attribute__((ext_vector_type(8))) float v8f;

__device__ __forceinline__ float fsign(float v) {
    // jnp.sign semantics: -1 / 0 / +1
    return (float)((v > 0.0f) - (v < 0.0f));
}

// ---------------------------------------------------------------------------
// Pass 1: gate logits with WMMA f32 16x16x4.
// One 256-thread block (8 wave32) per 16-row tile of x (rows = b*T+t, stride D).
// Wave w reduces K slice [w*512, (w+1)*512) with 4 interleaved WMMA chains.
// B-matrix column 0 = wg_w chunk, column 1 = rg_w chunk (broadcast GEMV trick):
// D[m,0] = partial w-logit of row m, D[m,1] = partial r-logit.
// B loads are branch-free: every lane loads, unwanted columns are zeroed by a
// hoisted mask so no exec-mask save/restore appears in the hot loop.
// ---------------------------------------------------------------------------
__global__ void __launch_bounds__(256)
gates_kernel(const float* __restrict__ x,
             const float* __restrict__ wg_w, const float* __restrict__ wg_b,
             const float* __restrict__ rg_w, const float* __restrict__ rg_b,
             float* __restrict__ wgate, float* __restrict__ rgate) {
    __shared__ float part[8][2][16];

    const int tid  = threadIdx.x;
    const int wave = tid >> 5;
    const int lane = tid & 31;
    const int r0   = blockIdx.x * 16;   // global row base (b*T + t)
    const int m    = lane & 15;         // A-matrix row M for this lane
    const int hi   = lane >> 4;         // 0: holds K0,K1 ; 1: holds K2,K3
    const int col  = lane & 15;         // D-matrix column N for this lane

    // A layout (32-bit A 16x4, ISA 7.12.2): lanes 0-15 -> {K0,K1}, 16-31 -> {K2,K3}
    const float* arow = x + (size_t)(r0 + m) * kD + hi * 2;
    // B layout (4x16): lanes 0-15 -> {K0,K1} of column N=lane,
    //                  lanes 16-31 -> {K2,K3} of column N=lane-16.
    // Column 0 reads wg_w, column 1 reads rg_w, all other lanes read rg_w too
    // (harmless duplicate) and get zeroed by zmask.
    const float* wrow  = ((col == 0) ? wg_w : rg_w) + hi * 2;
    const float  zmask = (col < 2) ? 1.0f : 0.0f;

    const v8f z = {0.f, 0.f, 0.f, 0.f, 0.f, 0.f, 0.f, 0.f};
    v8f c0 = z, c1 = z, c2 = z, c3 = z;

    const int kbeg = wave * (kD / 8);          // 512-wide K slice per wave
    const int kend = kbeg + (kD / 8);
    for (int k0 = kbeg; k0 < kend; k0 += 16) {
#pragma unroll
        for (int u = 0; u < 4; ++u) {
            const int kk = k0 + 4 * u;
            v2f a;
            a.x = arow[kk];
            a.y = arow[kk + 1];
            v2f b;
            b.x = wrow[kk] * zmask;
            b.y = wrow[kk + 1] * zmask;
            // 8-arg pattern: (neg_a, A, neg_b, B, c_mod, C, reuse_a, reuse_b)
            if (u == 0)
                c0 = __builtin_amdgcn_wmma_f32_16x16x4_f32(false, a, false, b,
                                                           (short)0, c0, false, false);
            else if (u == 1)
                c1 = __builtin_amdgcn_wmma_f32_16x16x4_f32(false, a, false, b,
                                                           (short)0, c1, false, false);
            else if (u == 2)
                c2 = __builtin_amdgcn_wmma_f32_16x16x4_f32(false, a, false, b,
                                                           (short)0, c2, false, false);
            else
                c3 = __builtin_amdgcn_wmma_f32_16x16x4_f32(false, a, false, b,
                                                           (short)0, c3, false, false);
        }
    }
    const v8f c = (c0 + c1) + (c2 + c3);

    // C/D layout: VGPR r -> M=r (lanes 0-15) / M=8+r (lanes 16-31), N=lane%16.
    if (col < 2) {
        const int rowoff = hi * 8;
#pragma unroll
        for (int r = 0; r < 8; ++r) part[wave][col][rowoff + r] = c[r];
    }
    __syncthreads();

    if (tid < 32) {
        const int g   = tid >> 4;   // 0 = write gate, 1 = read gate
        const int row = tid & 15;
        float s = 0.0f;
#pragma unroll
        for (int w = 0; w < 8; ++w) s += part[w][g][row];
        s += g ? rg_b[0] : wg_b[0];
        const float gate = 1.0f / (1.0f + expf(-s));
        (g ? rgate : wgate)[r0 + row] = gate;
    }
}

// ---------------------------------------------------------------------------
// Pass 2a: per-chunk sums of gated facts. sign(role_write) is +-1/0, so it is
// factored out of the t-loop (exact). grid = (D/256, NCHUNK, B).
// ---------------------------------------------------------------------------
__global__ void __launch_bounds__(256)
chunksum_kernel(const float* __restrict__ x,
                const float* __restrict__ role_write,
                const float* __restrict__ wgate,
                float* __restrict__ csum) {
    const int d     = blockIdx.x * 256 + threadIdx.x;
    const int chunk = blockIdx.y;
    const int b     = blockIdx.z;
    const size_t rowbase = (size_t)b * kT + (size_t)chunk * kChunk;
    const float* xp = x + rowbase * kD + d;
    const float* wg = wgate + rowbase;

    float s = 0.0f;
    for (int t = 0; t < kChunk; ++t)
        s += fsign(xp[(size_t)t * kD]) * wg[t];

    csum[((size_t)b * kNChunk + chunk) * kD + d] = s * fsign(role_write[d]);
}

// ---------------------------------------------------------------------------
// Pass 2b: exclusive prefix over the 32 chunks, one thread per (b,d) column.
// ---------------------------------------------------------------------------
__global__ void __launch_bounds__(256)
prefix_kernel(float* __restrict__ csum) {
    const int i = blockIdx.x * 256 + threadIdx.x;   // over B*D columns
    const int b = i / kD;
    const int d = i % kD;
    float run = 0.0f;
    for (int c = 0; c < kNChunk; ++c) {
        const size_t idx = ((size_t)b * kNChunk + c) * kD + d;
        const float v = csum[idx];
        csum[idx] = run;     // exclusive prefix, in place
        run += v;
    }
}

// ---------------------------------------------------------------------------
// Pass 2c: final in-chunk scan + read path + output. grid = (D/256, NCHUNK, B).
// ---------------------------------------------------------------------------
__global__ void __launch_bounds__(256)
output_kernel(const float* __restrict__ x,
              const float* __restrict__ role_write,
              const float* __restrict__ role_read,
              const float* __restrict__ wgate,
              const float* __restrict__ rgate,
              const float* __restrict__ csum,
              float* __restrict__ out) {
    const int d     = blockIdx.x * 256 + threadIdx.x;
    const int chunk = blockIdx.y;
    const int b     = blockIdx.z;
    const size_t rowbase = (size_t)b * kT + (size_t)chunk * kChunk;
    const float* xp = x + rowbase * kD + d;
    float*       op = out + rowbase * kD + d;
    const float* wg = wgate + rowbase;
    const float* rg = rgate + rowbase;

    const float srw = fsign(role_write[d]);
    const float srr = fsign(role_read[d]);
    float mem = csum[((size_t)b * kNChunk + chunk) * kD + d];

    for (int t = 0; t < kChunk; ++t) {
        const float xv = xp[(size_t)t * kD];
        const float s  = fsign(xv);
        mem += s * srw * wg[t];                       // causal bundle
        const float retrieved = fsign(mem) * s * srr; // unbind with read role
        op[(size_t)t * kD] = xv + rg[t] * retrieved;
    }
}

// ---------------------------------------------------------------------------
extern "C" void kernel_launch(void* const* d_in, const int* in_sizes, int n_in,
                              void* d_out, int out_size, void* d_ws, size_t ws_size,
                              hipStream_t stream) {
    (void)in_sizes; (void)n_in; (void)out_size; (void)ws_size;

    const float* x          = (const float*)d_in[0];
    const float* role_write = (const float*)d_in[1];
    const float* role_read  = (const float*)d_in[2];
    const float* wg_w       = (const float*)d_in[3];
    const float* wg_b       = (const float*)d_in[4];
    const float* rg_w       = (const float*)d_in[5];
    const float* rg_b       = (const float*)d_in[6];
    float* out = (float*)d_out;

    // Workspace layout (floats): [wgate: B*T][rgate: B*T][csum: B*NCHUNK*D]
    float* ws    = (float*)d_ws;
    float* wgate = ws;
    float* rgate = ws + kRows;
    float* csum  = ws + 2 * kRows;

    // Pass 1: 512 tiles of 16 rows, 8 waves each.
    gates_kernel<<<kRows / 16, 256, 0, stream>>>(x, wg_w, wg_b, rg_w, rg_b,
                                                 wgate, rgate);

    const dim3 g2(kD / 256, kNChunk, kB);
    chunksum_kernel<<<g2, 256, 0, stream>>>(x, role_write, wgate, csum);
    prefix_kernel<<<(kB * kD) / 256, 256, 0, stream>>>(csum);
    output_kernel<<<g2, 256, 0, stream>>>(x, role_write, role_read,
                                          wgate, rgate, csum, out);
}